// TranscriptionCRFModel_86079734546687
// MI455X (gfx1250) — compile-verified
//
#include <hip/hip_runtime.h>
#include <hip/hip_bf16.h>
#include <cstdint>

typedef __attribute__((ext_vector_type(8))) float v8f;
typedef __attribute__((ext_vector_type(2))) float v2f;

#define B_DIM  16
#define T_DIM  4096
#define KROOT  169
#define KBASS  13
#define KKEY   25

// ---------------------------------------------------------------------------
// GEMM: em[m, j] = sum_k X[m,k] * W[j,k] + bias[j]
// Full-f32 WMMA (v_wmma_f32_16x16x4_f32), exact-precision match to reference.
// One wave owns a 16-row M tile and iterates all N tiles, reusing the A
// fragment across N (X streamed exactly once; W stays cache-resident).
//   A layout (32-bit, 16x4): lane m in [0,16): K=0,1 ; lane m+16: K=2,3
//     => element e of lane: k = 4*kt + 2*h + e   (h = lane>>4)
//   B = W^T: lane's column j = nt*16 + (lane&15) is row j of W, same K split.
//   D: VGPR r -> row r + 8*h, col = lane&15.
// ---------------------------------------------------------------------------
template<int K>
__launch_bounds__(128)
__global__ void emit_gemm_wmma_f32(const float* __restrict__ X,
                                   const float* __restrict__ W,
                                   const float* __restrict__ bias,
                                   float* __restrict__ em)
{
    constexpr int NTILES = (K + 15) / 16;
    constexpr int KFULL  = K / 4;           // unguarded k-steps of 4
    constexpr int KREM   = K - KFULL * 4;   // 0..3 tail

    const int lane = threadIdx.x & 31;
    const int wave = threadIdx.x >> 5;
    const int h    = lane >> 4;
    const int mr   = lane & 15;

    const int mtile = blockIdx.x * 4 + wave;
    const int row0  = mtile * 16;
    const float* xrow = X + (size_t)(row0 + mr) * K;

    v8f zero = {};
    v8f acc[NTILES];
#pragma unroll
    for (int nt = 0; nt < NTILES; ++nt) acc[nt] = zero;

    for (int kt = 0; kt < KFULL; ++kt) {
        const int kb = kt * 4 + 2 * h;
        v2f a;
        a[0] = xrow[kb + 0];
        a[1] = xrow[kb + 1];
#pragma unroll
        for (int nt = 0; nt < NTILES; ++nt) {
            const int colB = nt * 16 + mr;
            v2f b = {};
            if (colB < K) {
                const float* wrow = W + (size_t)colB * K;
                b[0] = wrow[kb + 0];
                b[1] = wrow[kb + 1];
            }
            acc[nt] = __builtin_amdgcn_wmma_f32_16x16x4_f32(
                false, a, false, b, (short)0, acc[nt], false, false);
        }
    }
    if constexpr (KREM > 0) {
        const int kb = KFULL * 4 + 2 * h;
        v2f a = {};
        if (kb + 0 < K) a[0] = xrow[kb + 0];
        if (kb + 1 < K) a[1] = xrow[kb + 1];
#pragma unroll
        for (int nt = 0; nt < NTILES; ++nt) {
            const int colB = nt * 16 + mr;
            v2f b = {};
            if (colB < K) {
                const float* wrow = W + (size_t)colB * K;
                if (kb + 0 < K) b[0] = wrow[kb + 0];
                if (kb + 1 < K) b[1] = wrow[kb + 1];
            }
            acc[nt] = __builtin_amdgcn_wmma_f32_16x16x4_f32(
                false, a, false, b, (short)0, acc[nt], false, false);
        }
    }

#pragma unroll
    for (int nt = 0; nt < NTILES; ++nt) {
        const int col = nt * 16 + mr;
        if (col < K) {
            const float bv = bias[col];
#pragma unroll
            for (int r = 0; r < 8; ++r) {
                const int m = row0 + r + 8 * h;
                em[(size_t)m * K + col] = acc[nt][r] + bv;
            }
        }
    }
}

// ---------------------------------------------------------------------------
// CRF NLL for one batch chain per block.  loss contribution = (denom-numer)/B
// Dynamic LDS: trans (K*K) + two alpha buffers (K each).
// ---------------------------------------------------------------------------
template<int K>
__global__ void crf_nll_kernel(const float* __restrict__ em,    // (B,T,K)
                               const int*   __restrict__ tags,  // (B,T)
                               const int*   __restrict__ mask,  // (B,T)
                               const float* __restrict__ start, // (K)
                               const float* __restrict__ trans, // (K,K)
                               const float* __restrict__ endv,  // (K)
                               float*       __restrict__ out)
{
    extern __shared__ float smem[];
    float* tr  = smem;                 // K*K
    float* al0 = smem + (size_t)K * K; // K
    float* al1 = al0 + K;              // K
    __shared__ float sred[256];
    __shared__ int   sredi[256];

    const int b   = blockIdx.x;
    const int tid = threadIdx.x;
    const int NT  = blockDim.x;        // 256
    const float* emb = em   + (size_t)b * T_DIM * K;
    const int*   tgb = tags + (size_t)b * T_DIM;
    const int*   mkb = mask + (size_t)b * T_DIM;

    for (int i = tid; i < K * K; i += NT) tr[i] = trans[i];
    if (tid < K) al0[tid] = start[tid] + emb[tid];
    __syncthreads();

    // ---- forward recursion: alpha_t[j] = LSE_i(alpha[i] + trans[i,j]) + em_t[j]
    float* cur = al0;
    float* nxt = al1;
    for (int t = 1; t < T_DIM; ++t) {
        float anew = 0.0f;
        if (tid < K) {
            float mx = -3.402823466e38f;
            for (int i = 0; i < K; ++i)
                mx = fmaxf(mx, cur[i] + tr[i * K + tid]);
            float s = 0.0f;
            for (int i = 0; i < K; ++i)
                s += __expf(cur[i] + tr[i * K + tid] - mx);
            const float nv = mx + __logf(s) + emb[(size_t)t * K + tid];
            anew = mkb[t] ? nv : cur[tid];
        }
        if (tid < K) nxt[tid] = anew;
        __syncthreads();
        float* tmp = cur; cur = nxt; nxt = tmp;
    }

    // ---- denom = logsumexp(alpha_T + end)
    const float x = (tid < K) ? (cur[tid] + endv[tid]) : -3.402823466e38f;
    sred[tid] = x;
    __syncthreads();
    for (int off = NT >> 1; off > 0; off >>= 1) {
        if (tid < off) sred[tid] = fmaxf(sred[tid], sred[tid + off]);
        __syncthreads();
    }
    const float gmax = sred[0];
    __syncthreads();
    sred[tid] = (tid < K) ? __expf(x - gmax) : 0.0f;
    __syncthreads();
    for (int off = NT >> 1; off > 0; off >>= 1) {
        if (tid < off) sred[tid] += sred[tid + off];
        __syncthreads();
    }
    const float denom = gmax + __logf(sred[0]);
    __syncthreads();

    // ---- numerator: gold-path score (parallel over t) + mask popcount
    float part = 0.0f;
    int   msum = 0;
    for (int t = tid; t < T_DIM; t += NT) {
        const int tg = tgb[t];
        msum += mkb[t];
        if (t == 0) {
            part += start[tg] + emb[tg];
        } else if (mkb[t]) {
            part += tr[tgb[t - 1] * K + tg] + emb[(size_t)t * K + tg];
        }
    }
    sred[tid]  = part;
    sredi[tid] = msum;
    __syncthreads();
    for (int off = NT >> 1; off > 0; off >>= 1) {
        if (tid < off) { sred[tid] += sred[tid + off]; sredi[tid] += sredi[tid + off]; }
        __syncthreads();
    }
    if (tid == 0) {
        const int   seq_end = sredi[0] - 1;
        const float numer   = sred[0] + endv[tgb[seq_end]];
        atomicAdd(out, (denom - numer) * (1.0f / (float)B_DIM));
    }
}

__global__ void zero_out_kernel(float* p)
{
    if (threadIdx.x == 0 && blockIdx.x == 0) p[0] = 0.0f;
}

// ---------------------------------------------------------------------------
extern "C" void kernel_launch(void* const* d_in, const int* in_sizes, int n_in,
                              void* d_out, int out_size, void* d_ws, size_t ws_size,
                              hipStream_t stream)
{
    const float* root_logits = (const float*)d_in[0];
    const float* bass_logits = (const float*)d_in[1];
    const float* key_logits  = (const float*)d_in[2];
    const int*   root_labels = (const int*)  d_in[3];
    const int*   bass_labels = (const int*)  d_in[4];
    const int*   key_labels  = (const int*)  d_in[5];
    const int*   mask        = (const int*)  d_in[6];
    const float* W_root      = (const float*)d_in[7];
    const float* b_root      = (const float*)d_in[8];
    const float* W_bass      = (const float*)d_in[9];
    const float* b_bass      = (const float*)d_in[10];
    const float* W_key       = (const float*)d_in[11];
    const float* b_key       = (const float*)d_in[12];
    const float* start_root  = (const float*)d_in[13];
    const float* trans_root  = (const float*)d_in[14];
    const float* end_root    = (const float*)d_in[15];
    const float* start_bass  = (const float*)d_in[16];
    const float* trans_bass  = (const float*)d_in[17];
    const float* end_bass    = (const float*)d_in[18];
    const float* start_key   = (const float*)d_in[19];
    const float* trans_key   = (const float*)d_in[20];
    const float* end_key     = (const float*)d_in[21];
    float* out = (float*)d_out;

    const int M = B_DIM * T_DIM;   // 65536 rows

    // workspace layout (em buffers ~54 MB total: L2-resident on MI455X)
    char*  ws  = (char*)d_ws;
    size_t off = 0;
    auto carve = [&](size_t bytes) {
        void* p = ws + off;
        off = (off + bytes + 255) & ~(size_t)255;
        return p;
    };
    float* em_root = (float*)carve((size_t)M * KROOT * sizeof(float));
    float* em_bass = (float*)carve((size_t)M * KBASS * sizeof(float));
    float* em_key  = (float*)carve((size_t)M * KKEY  * sizeof(float));

    zero_out_kernel<<<1, 32, 0, stream>>>(out);

    // Pass 1: f32 WMMA GEMMs (em = logits @ W^T + b); X streamed once.
    dim3 blk(128);
    dim3 gm((unsigned)(M / 64), 1);   // 4 waves/block, 16 rows/wave
    emit_gemm_wmma_f32<KROOT><<<gm, blk, 0, stream>>>(root_logits, W_root, b_root, em_root);
    emit_gemm_wmma_f32<KBASS><<<gm, blk, 0, stream>>>(bass_logits, W_bass, b_bass, em_bass);
    emit_gemm_wmma_f32<KKEY> <<<gm, blk, 0, stream>>>(key_logits,  W_key,  b_key,  em_key);

    // Pass 2: CRF NLL per batch chain (em still hot in L2)
    const size_t sh_root = (size_t)(KROOT * KROOT + 2 * KROOT) * sizeof(float);
    const size_t sh_bass = (size_t)(KBASS * KBASS + 2 * KBASS) * sizeof(float);
    const size_t sh_key  = (size_t)(KKEY  * KKEY  + 2 * KKEY ) * sizeof(float);
    crf_nll_kernel<KROOT><<<B_DIM, 256, sh_root, stream>>>(
        em_root, root_labels, mask, start_root, trans_root, end_root, out);
    crf_nll_kernel<KBASS><<<B_DIM, 256, sh_bass, stream>>>(
        em_bass, bass_labels, mask, start_bass, trans_bass, end_bass, out);
    crf_nll_kernel<KKEY><<<B_DIM, 256, sh_key, stream>>>(
        em_key, key_labels, mask, start_key, trans_key, end_key, out);
}